// CustomLoss_19061064859882
// MI455X (gfx1250) — compile-verified
//
#include <hip/hip_runtime.h>
#include <math.h>

#define NCLS 128
#define EPSF 1e-9f

typedef int v4i __attribute__((ext_vector_type(4)));
typedef __attribute__((address_space(1))) v4i* gp_t;   // global int4*
typedef __attribute__((address_space(3))) v4i* lp_t;   // LDS int4*

// Each lane async-copies its 16 bytes of the 512-byte row into LDS.
// One instruction per wave -> ASYNCcnt += 1 for the wave.
__device__ __forceinline__ void async_load_row(const float* __restrict__ gsrc,
                                               float* ldst, int lane) {
    const float* g = gsrc + lane * 4;
    float*       l = ldst + lane * 4;
    __builtin_amdgcn_global_load_async_to_lds_b128(
        (gp_t)(void*)const_cast<float*>(g), (lp_t)(void*)l, 0, 0);
}

__global__ __launch_bounds__(256) void loss_main(
    const float* __restrict__ yp, const int* __restrict__ yt,
    double* __restrict__ acc,          // [0]=ce_sum, [1]=A0, [2+c]=L[c]
    unsigned* __restrict__ hist,       // n[c]
    int nrows)
{
    __shared__ __align__(16) float tile[8][2][NCLS];   // per-wave double buffer
    __shared__ float    L_blk[NCLS];
    __shared__ unsigned n_blk[NCLS];
    __shared__ float    ce_blk, A0_blk;

    const int tid  = threadIdx.x;
    const int lane = tid & 31;
    const int wave = tid >> 5;

    for (int i = tid; i < NCLS; i += 256) { L_blk[i] = 0.0f; n_blk[i] = 0u; }
    if (tid == 0) { ce_blk = 0.0f; A0_blk = 0.0f; }
    __syncthreads();

    const int wgid   = blockIdx.x * 8 + wave;   // global wave id
    const int stride = gridDim.x * 8;

    int row = wgid;
    int buf = 0;
    if (row < nrows) async_load_row(yp + (size_t)row * NCLS, &tile[wave][0][0], lane);

    for (; row < nrows; row += stride) {
        const int nxt = row + stride;
        if (nxt < nrows) {   // wave-uniform branch
            async_load_row(yp + (size_t)nxt * NCLS, &tile[wave][buf ^ 1][0], lane);
            __builtin_amdgcn_s_wait_asynccnt(1);   // current buffer landed
        } else {
            __builtin_amdgcn_s_wait_asynccnt(0);
        }

        float4 v = *(const float4*)&tile[wave][buf][lane * 4];   // ds_load_b128

        // row max
        float m = fmaxf(fmaxf(v.x, v.y), fmaxf(v.z, v.w));
        #pragma unroll
        for (int off = 16; off > 0; off >>= 1)
            m = fmaxf(m, __shfl_xor(m, off, 32));

        // sum of exp
        float ssum = expf(v.x - m) + expf(v.y - m) + expf(v.z - m) + expf(v.w - m);
        #pragma unroll
        for (int off = 16; off > 0; off >>= 1)
            ssum += __shfl_xor(ssum, off, 32);

        const int c = yt[row];                 // wave-uniform
        const int srcLane = c >> 2, sub = c & 3;
        float xl = (sub == 0) ? v.x : (sub == 1) ? v.y : (sub == 2) ? v.z : v.w;
        float xt = __shfl(xl, srcLane, 32);    // logit at true class
        float x0 = __shfl(v.x, 0, 32);         // logit of class 0

        if (lane == 0) {
            float lse    = m + logf(ssum);
            float logp_t = xt - lse;
            atomicAdd(&ce_blk, -logp_t);                        // ds_add_f32
            float p1 = expf(x0 - lse);                          // softmax p[:,0]
            if (c == 0) atomicAdd(&A0_blk, logf(p1 + EPSF));
            else        atomicAdd(&L_blk[c], logf(1.0f - p1 + EPSF));
            atomicAdd(&n_blk[c], 1u);                           // ds_add_u32
        }
        buf ^= 1;
    }
    __syncthreads();

    // flush block partials (double accumulators in global for accuracy)
    for (int i = tid; i < NCLS; i += 256) {
        if (n_blk[i])            atomicAdd(&hist[i], n_blk[i]);
        if (i >= 1 && L_blk[i] != 0.0f) atomicAdd(&acc[2 + i], (double)L_blk[i]);
    }
    if (tid == 0) {
        atomicAdd(&acc[0], (double)ce_blk);
        atomicAdd(&acc[1], (double)A0_blk);
    }
}

__global__ void loss_init(double* acc, unsigned* hist) {
    int t = threadIdx.x;
    if (t < 130) acc[t] = 0.0;
    if (t < NCLS) hist[t] = 0u;
}

__global__ void loss_final(const double* __restrict__ acc,
                           const unsigned* __restrict__ hist,
                           float* __restrict__ out, int nrows)
{
    __shared__ double red[4];
    const int tid = threadIdx.x;   // 128 threads
    const double denom = (double)nrows - (double)hist[0];
    double term = 0.0;
    if (tid >= 1) {
        double s = 0.5 * (1.0 - (double)hist[tid] / denom);  // BETA = 0.5
        term = s * acc[2 + tid];
    }
    #pragma unroll
    for (int off = 16; off > 0; off >>= 1)
        term += __shfl_down(term, off, 32);
    if ((tid & 31) == 0) red[tid >> 5] = term;
    __syncthreads();
    if (tid == 0) {
        double penalty = 0.5 * acc[1] + red[0] + red[1] + red[2] + red[3]; // ALPHA=0.5
        double ce = acc[0] / (double)nrows;
        out[0] = (float)(ce - penalty / (double)nrows);
    }
}

extern "C" void kernel_launch(void* const* d_in, const int* in_sizes, int n_in,
                              void* d_out, int out_size, void* d_ws, size_t ws_size,
                              hipStream_t stream) {
    const float* yp = (const float*)d_in[0];
    const int*   yt = (const int*)d_in[1];
    const int nrows = in_sizes[0] / NCLS;          // 262144

    double*   acc  = (double*)d_ws;                             // 130 doubles
    unsigned* hist = (unsigned*)((char*)d_ws + 130 * sizeof(double)); // 128 uints
    float*    out  = (float*)d_out;

    loss_init<<<1, 256, 0, stream>>>(acc, hist);
    loss_main<<<1024, 256, 0, stream>>>(yp, yt, acc, hist, nrows);
    loss_final<<<1, 128, 0, stream>>>(acc, hist, out, nrows);
}